// _ProposalLayer_63393717289576
// MI455X (gfx1250) — compile-verified
//
#include <hip/hip_runtime.h>
#include <hip/hip_bf16.h>
#include <stdint.h>

typedef float v2f __attribute__((ext_vector_type(2)));
typedef float v8f __attribute__((ext_vector_type(8)));

#define BATCH   16
#define NANCH   135000
#define NBUCK   2048
#define CAP     16384      // candidate capacity per batch (>= ~8000 worst case)
#define TOPK    6000       // PRE_NMS
#define POST    300        // POST_NMS
#define HH      100
#define WW      150
#define NMS_T   0.7f

// generate_anchors(16, [0.5,1,2], [8,16,32]) precomputed (ratio-major, scale-minor)
__device__ __constant__ float ANCH[9][4] = {
    {-84.f,  -40.f,  99.f,  55.f}, {-176.f,  -88.f, 191.f, 103.f}, {-360.f, -184.f, 375.f, 199.f},
    {-56.f,  -56.f,  71.f,  71.f}, {-120.f, -120.f, 135.f, 135.f}, {-248.f, -248.f, 263.f, 263.f},
    {-36.f,  -80.f,  51.f,  95.f}, { -80.f, -168.f,  95.f, 183.f}, {-168.f, -344.f, 183.f, 359.f}};

__global__ void fill_u32(unsigned* p, int n, unsigned v) {
    int i = blockIdx.x * 256 + threadIdx.x;
    if (i < n) p[i] = v;
}

// Monotone float->uint key; per-batch 2048-bucket histogram of key>>21.
__global__ void keys_hist(const float* __restrict__ scores,
                          unsigned* __restrict__ keys, unsigned* __restrict__ hist) {
    __shared__ unsigned hc[NBUCK];
    int tid = threadIdx.x;
    for (int i = tid; i < NBUCK; i += 256) hc[i] = 0;
    __syncthreads();
    int b = blockIdx.y;
    int n = blockIdx.x * 256 + tid;
    if (n < NANCH) {
        int a = n % 9, hw = n / 9, w = hw % WW, h = hw / WW;
        float s = scores[(((size_t)b * 18 + 9 + a) * HH + h) * WW + w];
        unsigned bits = __float_as_uint(s);
        unsigned key = (bits & 0x80000000u) ? ~bits : (bits | 0x80000000u);
        keys[(size_t)b * NANCH + n] = key;
        atomicAdd(&hc[key >> 21], 1u);
    }
    __syncthreads();
    for (int i = tid; i < NBUCK; i += 256) {
        unsigned v = hc[i];
        if (v) atomicAdd(&hist[b * NBUCK + i], v);
    }
}

// One wave32 per batch: find threshold bucket T s.t. count(bucket>=T) >= TOPK.
// Final 16-bin suffix sums via chained V_WMMA_F32_16X16X4_F32 (U-triangular x vec).
__global__ void scan_hist(const unsigned* __restrict__ hist, unsigned* __restrict__ thresh) {
    int b = blockIdx.x;
    int lane = threadIdx.x;
    __shared__ float csum[128];
    __shared__ float harr[16];
    __shared__ float s16[16];
    const unsigned* h = hist + b * NBUCK;
    for (int c = lane; c < 128; c += 32) {
        float s = 0.f;
        for (int t = 0; t < 16; ++t) s += (float)h[c * 16 + t];
        csum[c] = s;
    }
    __syncthreads();
    // uniform serial scan over 128 chunk sums from the top
    float acc = 0.f, above = 0.f;
    int C = 0;
    for (int c = 127; c >= 0; --c) {
        float na = acc + csum[c];
        if (na >= (float)TOPK) { C = c; above = acc; break; }
        acc = na;
    }
    if (lane < 16) harr[lane] = (float)h[C * 16 + lane];
    __syncthreads();
    // WMMA suffix scan of the 16 bins in chunk C:
    // D = U * v, U upper-triangular ones (16x16), v replicated across B columns.
    int row = lane & 15;
    int khalf = (lane >> 4) * 2;             // A/B layout: lanes16-31 hold K=2,3
    v8f accv = {0.f, 0.f, 0.f, 0.f, 0.f, 0.f, 0.f, 0.f};
    for (int kk = 0; kk < 4; ++kk) {
        int k0 = kk * 4 + khalf;
        v2f a, bm;
        a.x = (k0 >= row) ? 1.f : 0.f;
        a.y = (k0 + 1 >= row) ? 1.f : 0.f;
        bm.x = harr[k0];
        bm.y = harr[k0 + 1];
        accv = __builtin_amdgcn_wmma_f32_16x16x4_f32(false, a, false, bm,
                                                     (short)0, accv, false, false);
    }
    // D column 0: lane0 holds M=0..7, lane16 holds M=8..15
    if (lane == 0)  for (int r = 0; r < 8; ++r) s16[r] = accv[r];
    if (lane == 16) for (int r = 0; r < 8; ++r) s16[8 + r] = accv[r];
    __syncthreads();
    int T = 0;
    for (int i = 15; i >= 0; --i)
        if (above + s16[i] >= (float)TOPK) { T = C * 16 + i; break; }
    if (lane == 0) thresh[b] = (unsigned)T;
}

__global__ void compact(const unsigned* __restrict__ keys, const unsigned* __restrict__ thresh,
                        unsigned* __restrict__ cnt, unsigned long long* __restrict__ sel) {
    int b = blockIdx.y;
    int n = blockIdx.x * 256 + threadIdx.x;
    if (n >= NANCH) return;
    unsigned key = keys[(size_t)b * NANCH + n];
    if ((key >> 21) >= thresh[b]) {
        unsigned pos = atomicAdd(&cnt[b], 1u);
        if (pos < CAP)
            sel[(size_t)b * CAP + pos] = (((unsigned long long)(~key)) << 32) | (unsigned)n;
    }
}

// One block per batch: async copy -> LDS, bitonic sort, gather boxes, greedy NMS.
__global__ __launch_bounds__(1024) void sort_nms(const unsigned long long* __restrict__ sel,
                                                 const float* __restrict__ deltas,
                                                 const float* __restrict__ im_info,
                                                 float* __restrict__ out) {
    // 131072 B: sort buffer (16384 x u64) then reused as 6000 float4 boxes (96000 B)
    __shared__ __align__(16) float smem[CAP * 2];
    __shared__ unsigned char supp[TOPK];
    __shared__ int s_cur;
    unsigned long long* sortbuf = reinterpret_cast<unsigned long long*>(smem);

    int b = blockIdx.x, tid = threadIdx.x;

    // ---- Phase 1: CDNA5 async global->LDS bulk copy of candidate list ----
    {
        unsigned long long gbase = (unsigned long long)(const void*)(sel + (size_t)b * CAP);
        unsigned lbase = (unsigned)(unsigned long long)(void*)&smem[0];
        for (int t = tid; t < (CAP * 8) / 16; t += 1024) {
            unsigned loff = lbase + (unsigned)t * 16u;
            unsigned long long ga = gbase + (unsigned long long)t * 16ull;
            asm volatile("global_load_async_to_lds_b128 %0, %1, off"
                         :: "v"(loff), "v"(ga) : "memory");
        }
        asm volatile("s_wait_asynccnt 0" ::: "memory");
    }
    __syncthreads();

    // ---- Phase 2: bitonic sort ascending ((~key)<<32 | idx) ----
    for (int k = 2; k <= CAP; k <<= 1) {
        for (int j = k >> 1; j > 0; j >>= 1) {
            for (int i = tid; i < CAP; i += 1024) {
                int ixj = i ^ j;
                if (ixj > i) {
                    unsigned long long va = sortbuf[i], vb = sortbuf[ixj];
                    bool up = ((i & k) == 0);
                    if ((va > vb) == up) { sortbuf[i] = vb; sortbuf[ixj] = va; }
                }
            }
            __syncthreads();
        }
    }

    // ---- Phase 3: stage top-6000 anchor indices into registers ----
    unsigned myn[8];
    int mc = 0;
    for (int i = tid; i < TOPK; i += 1024) myn[mc++] = (unsigned)(sortbuf[i] & 0xFFFFFFFFull);
    __syncthreads();   // all reads done before LDS reuse

    // ---- Phase 4: recompute proposals (delta transform + clip) into LDS ----
    float hmax = im_info[b * 3 + 0] - 1.f;
    float wmax = im_info[b * 3 + 1] - 1.f;
    float4* boxes = reinterpret_cast<float4*>(smem);
    mc = 0;
    for (int i = tid; i < TOPK; i += 1024) {
        unsigned n = myn[mc++];
        int a = (int)(n % 9), hw = (int)(n / 9), w = hw % WW, h = hw / WW;
        float ax1 = ANCH[a][0] + w * 16.f, ay1 = ANCH[a][1] + h * 16.f;
        float ax2 = ANCH[a][2] + w * 16.f, ay2 = ANCH[a][3] + h * 16.f;
        const float* dp = deltas + (((size_t)b * 36 + a * 4) * HH + h) * WW + w;
        float d0 = dp[0], d1 = dp[HH * WW], d2 = dp[2 * HH * WW], d3 = dp[3 * HH * WW];
        float aw = ax2 - ax1 + 1.f, ah = ay2 - ay1 + 1.f;
        float cx = ax1 + 0.5f * aw, cy = ay1 + 0.5f * ah;
        float pcx = d0 * aw + cx, pcy = d1 * ah + cy;
        float pw = expf(d2) * aw, ph = expf(d3) * ah;
        float x1 = pcx - 0.5f * pw, y1 = pcy - 0.5f * ph;
        float x2 = pcx + 0.5f * pw, y2 = pcy + 0.5f * ph;
        x1 = fminf(fmaxf(x1, 0.f), wmax);
        y1 = fminf(fmaxf(y1, 0.f), hmax);
        x2 = fminf(fmaxf(x2, 0.f), wmax);
        y2 = fminf(fmaxf(y2, 0.f), hmax);
        boxes[i] = make_float4(x1, y1, x2, y2);
        supp[i] = 0;
    }
    __syncthreads();

    // ---- Phase 5: greedy NMS, 300 sequential picks ----
    float* ob = out + (size_t)b * POST * 5;
    int kept = 0, cur = 0;
    while (kept < POST) {
        if (tid == 0) {
            int c = cur;
            while (c < TOPK && supp[c]) ++c;
            s_cur = c;
        }
        __syncthreads();
        cur = s_cur;
        if (cur >= TOPK) break;
        float4 bx = boxes[cur];
        if (tid == 0) {
            float* r = ob + kept * 5;
            r[0] = (float)b; r[1] = bx.x; r[2] = bx.y; r[3] = bx.z; r[4] = bx.w;
            supp[cur] = 1;
        }
        float areaI = (bx.z - bx.x + 1.f) * (bx.w - bx.y + 1.f);
        for (int j = cur + 1 + tid; j < TOPK; j += 1024) {
            if (!supp[j]) {
                float4 bj = boxes[j];
                float xx1 = fmaxf(bx.x, bj.x), yy1 = fmaxf(bx.y, bj.y);
                float xx2 = fminf(bx.z, bj.z), yy2 = fminf(bx.w, bj.w);
                float iw = fmaxf(0.f, xx2 - xx1 + 1.f), ih = fmaxf(0.f, yy2 - yy1 + 1.f);
                float inter = iw * ih;
                float areaJ = (bj.z - bj.x + 1.f) * (bj.w - bj.y + 1.f);
                if (inter / (areaI + areaJ - inter) > NMS_T) supp[j] = 1;
            }
        }
        ++kept; ++cur;
        __syncthreads();
    }
    // zero-fill remaining rows (reference writes zeros when out of valid boxes)
    for (int z = kept * 5 + tid; z < POST * 5; z += 1024) ob[z] = 0.f;
}

extern "C" void kernel_launch(void* const* d_in, const int* in_sizes, int n_in,
                              void* d_out, int out_size, void* d_ws, size_t ws_size,
                              hipStream_t stream) {
    const float* scores  = (const float*)d_in[0];
    const float* deltas  = (const float*)d_in[1];
    const float* im_info = (const float*)d_in[2];
    float* out = (float*)d_out;

    char* ws = (char*)d_ws;
    size_t off = 0;
    auto take = [&](size_t bytes) -> void* {
        void* p = ws + off;
        off += (bytes + 255) & ~(size_t)255;
        return p;
    };
    unsigned* keys            = (unsigned*)take((size_t)BATCH * NANCH * 4);
    unsigned* hist            = (unsigned*)take((size_t)BATCH * NBUCK * 4);
    unsigned* thresh          = (unsigned*)take((size_t)BATCH * 4);
    unsigned* cnt             = (unsigned*)take((size_t)BATCH * 4);
    unsigned long long* sel   = (unsigned long long*)take((size_t)BATCH * CAP * 8);
    (void)ws_size; (void)n_in; (void)in_sizes; (void)out_size;

    int nh = BATCH * NBUCK;
    fill_u32<<<(nh + 255) / 256, 256, 0, stream>>>(hist, nh, 0u);
    fill_u32<<<1, 256, 0, stream>>>(cnt, BATCH, 0u);
    int nsel = BATCH * CAP * 2;
    fill_u32<<<(nsel + 255) / 256, 256, 0, stream>>>((unsigned*)sel, nsel, 0xFFFFFFFFu);

    dim3 grid((NANCH + 255) / 256, BATCH);
    keys_hist<<<grid, 256, 0, stream>>>(scores, keys, hist);
    scan_hist<<<BATCH, 32, 0, stream>>>(hist, thresh);
    compact<<<grid, 256, 0, stream>>>(keys, thresh, cnt, sel);
    sort_nms<<<BATCH, 1024, 0, stream>>>(sel, deltas, im_info, out);
}